// ASVT_9500467658791
// MI455X (gfx1250) — compile-verified
//
#include <hip/hip_runtime.h>

typedef __attribute__((ext_vector_type(2))) float v2f;
typedef __attribute__((ext_vector_type(8))) float v8f;

#define DIM      256
#define DQK      32
#define NBATCH   8
#define LMAXPAD  3072
#define QTILE    128                       // 8 waves * 16 query rows
#define TILESMAX ((LMAXPAD + QTILE - 1) / QTILE)
#define KCHUNK   32
#define K2STR    18                        // v2f stride (36 floats, 144B: B128-aligned, conflict-free)
#define VTSTR    17                        // v2f stride (34 floats): transposed V, conflict-free
#define P2STR    9                         // v2f stride (18 floats)
#define BN_EPS   1e-5f

__device__ __forceinline__ v8f wmma4(v2f a, v2f b, v8f c) {
  // V_WMMA_F32_16X16X4_F32 : D = A(16x4) x B(4x16) + C(16x16), fp32 throughout
  return __builtin_amdgcn_wmma_f32_16x16x4_f32(false, a, false, b, (short)0, c,
                                               false, false);
}

// CDNA5 async copy: global -> LDS, 16B per lane, tracked by ASYNCcnt
__device__ __forceinline__ void async_load_b128(unsigned lds_off, const float* g) {
  asm volatile("global_load_async_to_lds_b128 %0, %1, off"
               :: "v"(lds_off), "v"(g) : "memory");
}
__device__ __forceinline__ void wait_async0() {
  asm volatile("s_wait_asynccnt 0x0" ::: "memory");
}

// ---------------------------------------------------------------------------
// setup: segment starts via binary search on sorted bids; zero BN accumulators
// ---------------------------------------------------------------------------
__global__ void ASVT_setup_kernel(const int* __restrict__ bids, int* __restrict__ starts,
                                  float* __restrict__ sum, float* __restrict__ sumsq, int N) {
  int t = threadIdx.x;
  if (t < DIM) sum[t] = 0.0f;
  else if (t < 2 * DIM) sumsq[t - DIM] = 0.0f;
  if (t <= NBATCH) {
    if (t == NBATCH) {
      starts[NBATCH] = N;
    } else {
      int lo = 0, hi = N;
      while (lo < hi) { int mid = (lo + hi) >> 1; if (bids[mid] < t) lo = mid + 1; else hi = mid; }
      starts[t] = lo;
    }
  }
}

// ---------------------------------------------------------------------------
// qkv: fused [N,256] @ [256, 32|32|256] ; each wave -> 16 rows x 64 cols.
// cg==0 handles Wq|Wk (ldw=32, compile-time); cg 1..4 handle Wv (ldw=256).
// ---------------------------------------------------------------------------
__global__ __launch_bounds__(256) void ASVT_qkv_kernel(
    const float* __restrict__ feat, const float* __restrict__ Wq,
    const float* __restrict__ Wk, const float* __restrict__ Wv,
    float* __restrict__ q, float* __restrict__ k, float* __restrict__ v, int N) {
  const int gw = blockIdx.x * 8 + (threadIdx.x >> 5);
  const int rt = gw / 5, cg = gw % 5;
  if (rt * 16 >= N) return;
  const int lane = threadIdx.x & 31;
  const int half = lane >> 4, l16 = lane & 15;
  const int k0 = half * 2, mbase = half * 8;

  const v2f* arow = (const v2f*)(feat + (size_t)(rt * 16 + l16) * DIM);
  v8f zero = {};
  v8f acc[4] = {zero, zero, zero, zero};

  if (cg == 0) {                      // q (cols 0..31) and k (cols 0..31)
    const float* bp[4] = { Wq + l16 + k0 * DQK,      Wq + 16 + l16 + k0 * DQK,
                           Wk + l16 + k0 * DQK,      Wk + 16 + l16 + k0 * DQK };
#pragma unroll
    for (int kk = 0; kk < 64; ++kk) {
      const v2f a = arow[kk * 2 + half];
#pragma unroll
      for (int c = 0; c < 4; ++c) {
        v2f b;
        b[0] = bp[c][kk * 4 * DQK];
        b[1] = bp[c][kk * 4 * DQK + DQK];
        acc[c] = wmma4(a, b, acc[c]);
      }
    }
#pragma unroll
    for (int c = 0; c < 4; ++c) {
      float* ob = ((c < 2) ? q : k) + (size_t)(rt * 16 + mbase) * DQK + (c & 1) * 16 + l16;
#pragma unroll
      for (int vv = 0; vv < 8; ++vv) ob[vv * DQK] = acc[c][vv];
    }
  } else {                            // v (cols (cg-1)*64 .. +63)
    const float* bp0 = Wv + (cg - 1) * 64 + l16 + k0 * DIM;
#pragma unroll
    for (int kk = 0; kk < 64; ++kk) {
      const v2f a = arow[kk * 2 + half];
#pragma unroll
      for (int c = 0; c < 4; ++c) {
        v2f b;
        b[0] = bp0[c * 16 + kk * 4 * DIM];
        b[1] = bp0[c * 16 + kk * 4 * DIM + DIM];
        acc[c] = wmma4(a, b, acc[c]);
      }
    }
    float* ob = v + (size_t)(rt * 16 + mbase) * DIM + (cg - 1) * 64 + l16;
#pragma unroll
    for (int c = 0; c < 4; ++c)
#pragma unroll
      for (int vv = 0; vv < 8; ++vv) ob[vv * DIM + c * 16] = acc[c][vv];
  }
}

// ---------------------------------------------------------------------------
// attn: per-segment flash attention. Block = 128 queries of one segment.
// K staged via async-to-LDS (B128); V staged K-major so P@V B-frags are b64.
// ---------------------------------------------------------------------------
__global__ __launch_bounds__(256) void ASVT_attn_kernel(
    const float* __restrict__ q, const float* __restrict__ k, const float* __restrict__ v,
    const int* __restrict__ starts, float* __restrict__ r, int N) {
  __shared__ v2f k2[KCHUNK][K2STR];        // [key][kdim pair], stride 144B
  __shared__ v2f vt[DIM][VTSTR];           // [dim col][key pair] (transposed V)
  __shared__ v2f p2[8][16][P2STR];         // per-wave P tile [m][key pair]

  const int b    = blockIdx.x / TILESMAX;
  const int tile = blockIdx.x % TILESMAX;
  const int s0 = starts[b];
  const int L  = starts[b + 1] - s0;
  if (tile * QTILE >= L) return;           // uniform per block

  const int tid = threadIdx.x;
  const int wave = tid >> 5, lane = tid & 31;
  const int half = lane >> 4, l16 = lane & 15;
  const int k0 = half * 2, mbase = half * 8;

  float* k2f = (float*)&k2[0][0];
  float* vtf = (float*)&vt[0][0];
  float* p2f = (float*)&p2[0][0][0];

  // Q fragments for this wave's 16 rows (clamped for tail rows; not stored)
  const int qrow_local = tile * QTILE + wave * 16 + l16;
  const v2f* qv = (const v2f*)(q + (size_t)(s0 + min(qrow_local, L - 1)) * DQK);
  v2f qa[8];
#pragma unroll
  for (int kk = 0; kk < 8; ++kk) qa[kk] = qv[kk * 2 + half];

  v8f zero = {};
  v8f o[16];
#pragma unroll
  for (int t = 0; t < 16; ++t) o[t] = zero;
  float m_run[8], l_run[8];
#pragma unroll
  for (int i = 0; i < 8; ++i) { m_run[i] = -3.0e38f; l_run[i] = 0.0f; }

  // per-thread staging coordinates
  const int kr = tid >> 3, kc4 = tid & 7;    // K: 32 rows x 8 float4
  const unsigned klds = (unsigned)(size_t)(void*)&k2f[kr * (2 * K2STR) + kc4 * 4];

  const int nchunks = (L + KCHUNK - 1) / KCHUNK;
  for (int ch = 0; ch < nchunks; ++ch) {
    const int kbase = ch * KCHUNK;
    // --- K chunk: one async B128 per thread, straight to LDS ---
    async_load_b128(klds, k + (size_t)min(s0 + kbase + kr, N - 1) * DQK + kc4 * 4);
    // --- V chunk: 32x256 loaded row-major, stored K-major (transposed) ---
#pragma unroll
    for (int i = 0; i < 8; ++i) {
      const int idx = tid + i * 256;
      const int vr = idx >> 6, vc4 = idx & 63;
      const float4 val =
          *(const float4*)(v + (size_t)min(s0 + kbase + vr, N - 1) * DIM + vc4 * 4);
      vtf[(vc4 * 4 + 0) * (2 * VTSTR) + vr] = val.x;
      vtf[(vc4 * 4 + 1) * (2 * VTSTR) + vr] = val.y;
      vtf[(vc4 * 4 + 2) * (2 * VTSTR) + vr] = val.z;
      vtf[(vc4 * 4 + 3) * (2 * VTSTR) + vr] = val.w;
    }
    if (ch + 1 < nchunks)                    // warm L2/WGP$ for next chunk
      __builtin_prefetch(v + (size_t)min(s0 + kbase + KCHUNK + (tid >> 6), N - 1) * DIM
                             + (tid & 63) * 4, 0, 0);
    wait_async0();
    __syncthreads();

#pragma unroll
    for (int sub = 0; sub < 2; ++sub) {
      // S = Q(16x32) @ Kc^T(32x16) -- B frag is one ds_load_b64
      v8f s = zero;
#pragma unroll
      for (int kk = 0; kk < 8; ++kk)
        s = wmma4(qa[kk], k2[sub * 16 + l16][kk * 2 + half], s);
      // mask invalid keys (this lane's key column = l16)
      if (kbase + sub * 16 + l16 >= L) {
#pragma unroll
        for (int vv = 0; vv < 8; ++vv) s[vv] = -1.0e30f;
      }
      // online softmax: rows live across the 16 lanes of each half-wave
      float scale[8];
#pragma unroll
      for (int vv = 0; vv < 8; ++vv) {
        float mx = s[vv];
        mx = fmaxf(mx, __shfl_xor(mx, 1, 32));
        mx = fmaxf(mx, __shfl_xor(mx, 2, 32));
        mx = fmaxf(mx, __shfl_xor(mx, 4, 32));
        mx = fmaxf(mx, __shfl_xor(mx, 8, 32));
        const float mnew = fmaxf(m_run[vv], mx);
        scale[vv] = __expf(m_run[vv] - mnew);
        m_run[vv] = mnew;
        const float pr = __expf(s[vv] - mnew);
        s[vv] = pr;
        float sm = pr;
        sm += __shfl_xor(sm, 1, 32);
        sm += __shfl_xor(sm, 2, 32);
        sm += __shfl_xor(sm, 4, 32);
        sm += __shfl_xor(sm, 8, 32);
        l_run[vv] = l_run[vv] * scale[vv] + sm;
      }
      // rescale O, stage P (C-layout -> LDS -> A-layout; same-wave DS is in-order)
#pragma unroll
      for (int t = 0; t < 16; ++t)
#pragma unroll
        for (int vv = 0; vv < 8; ++vv) o[t][vv] *= scale[vv];
#pragma unroll
      for (int vv = 0; vv < 8; ++vv)
        p2f[wave * (16 * 2 * P2STR) + (mbase + vv) * (2 * P2STR) + l16] = s[vv];
      v2f pa[4];
#pragma unroll
      for (int kk = 0; kk < 4; ++kk) pa[kk] = p2[wave][l16][kk * 2 + half];
      // O += P(16x16) @ Vc(16x256) -- B frag is one ds_load_b64 (transposed V)
#pragma unroll
      for (int t = 0; t < 16; ++t)
#pragma unroll
        for (int kk = 0; kk < 4; ++kk)
          o[t] = wmma4(pa[kk], vt[t * 16 + l16][sub * 8 + kk * 2 + half], o[t]);
    }
    __syncthreads();                        // protect K/V LDS for next chunk
  }

  float inv_l[8];
#pragma unroll
  for (int vv = 0; vv < 8; ++vv) inv_l[vv] = 1.0f / l_run[vv];
  const int mloc = tile * QTILE + wave * 16 + mbase;
  float* rb = r + (size_t)(s0 + mloc) * DIM + l16;
#pragma unroll
  for (int vv = 0; vv < 8; ++vv) {
    if (mloc + vv < L) {
#pragma unroll
      for (int t = 0; t < 16; ++t)
        rb[vv * DIM + t * 16] = o[t][vv] * inv_l[vv];
    }
  }
}

// ---------------------------------------------------------------------------
// trans: t = r @ Wt, plus per-channel sum / sumsq for BatchNorm stats
// ---------------------------------------------------------------------------
__global__ __launch_bounds__(256) void ASVT_trans_kernel(
    const float* __restrict__ r, const float* __restrict__ Wt,
    float* __restrict__ t, float* __restrict__ sum, float* __restrict__ sumsq, int N) {
  const int gw = blockIdx.x * 8 + (threadIdx.x >> 5);
  const int rt = gw >> 2, cg = gw & 3;      // 256/64 = 4 col groups
  if (rt * 16 >= N) return;
  const int lane = threadIdx.x & 31;
  const int half = lane >> 4, l16 = lane & 15;
  const int k0 = half * 2, mbase = half * 8;

  const v2f* arow = (const v2f*)(r + (size_t)(rt * 16 + l16) * DIM);
  const float* bp0 = Wt + cg * 64 + l16 + k0 * DIM;
  v8f zero = {};
  v8f acc[4] = {zero, zero, zero, zero};
#pragma unroll
  for (int kk = 0; kk < 64; ++kk) {
    const v2f a = arow[kk * 2 + half];
#pragma unroll
    for (int c = 0; c < 4; ++c) {
      v2f b;
      b[0] = bp0[c * 16 + kk * 4 * DIM];
      b[1] = bp0[c * 16 + kk * 4 * DIM + DIM];
      acc[c] = wmma4(a, b, acc[c]);
    }
  }
  float* ob = t + (size_t)(rt * 16 + mbase) * DIM + cg * 64 + l16;
#pragma unroll
  for (int c = 0; c < 4; ++c) {
    const int col = cg * 64 + c * 16 + l16;
    float s1 = 0.0f, s2 = 0.0f;
#pragma unroll
    for (int vv = 0; vv < 8; ++vv) {
      const float x = acc[c][vv];
      ob[vv * DIM + c * 16] = x;
      s1 += x; s2 += x * x;
    }
    s1 += __shfl_xor(s1, 16, 32);
    s2 += __shfl_xor(s2, 16, 32);
    if (half == 0) {
      atomicAdd(&sum[col], s1);
      atomicAdd(&sumsq[col], s2);
    }
  }
}

// ---------------------------------------------------------------------------
// bn: out = feat + relu((t - mu) * rsqrt(var + eps) * gamma + beta)
// ---------------------------------------------------------------------------
__global__ __launch_bounds__(256) void ASVT_bn_kernel(
    const float* __restrict__ feat, const float* __restrict__ t,
    const float* __restrict__ sum, const float* __restrict__ sumsq,
    const float* __restrict__ gamma, const float* __restrict__ beta,
    float* __restrict__ out, int N) {
  const size_t total = (size_t)N * DIM;
  const float invN = 1.0f / (float)N;
  for (size_t idx = (size_t)blockIdx.x * blockDim.x + threadIdx.x; idx < total;
       idx += (size_t)gridDim.x * blockDim.x) {
    const int c = (int)(idx & (DIM - 1));
    const float mu = sum[c] * invN;
    const float var = sumsq[c] * invN - mu * mu;
    const float x = (t[idx] - mu) * rsqrtf(var + BN_EPS) * gamma[c] + beta[c];
    out[idx] = feat[idx] + fmaxf(x, 0.0f);
  }
}

// ---------------------------------------------------------------------------
extern "C" void kernel_launch(void* const* d_in, const int* in_sizes, int n_in,
                              void* d_out, int out_size, void* d_ws, size_t ws_size,
                              hipStream_t stream) {
  const float* feat  = (const float*)d_in[0];
  const int*   bids  = (const int*)d_in[1];
  const float* Wq    = (const float*)d_in[2];
  const float* Wk    = (const float*)d_in[3];
  const float* Wv    = (const float*)d_in[4];
  const float* Wt    = (const float*)d_in[5];
  const float* gamma = (const float*)d_in[6];
  const float* beta  = (const float*)d_in[7];
  float* out = (float*)d_out;

  const int N = in_sizes[0] / DIM;

  char* ws = (char*)d_ws;
  int*   starts = (int*)ws;                      // [NBATCH+1]
  float* sum    = (float*)(ws + 1024);           // [256]
  float* sumsq  = (float*)(ws + 2048);           // [256]
  float* q = (float*)(ws + 4096);                // [N,32]
  float* k = q + (size_t)N * DQK;                // [N,32]
  float* v = k + (size_t)N * DQK;                // [N,256]
  float* r = v + (size_t)N * DIM;                // [N,256]
  float* t = r + (size_t)N * DIM;                // [N,256]

  ASVT_setup_kernel<<<1, 512, 0, stream>>>(bids, starts, sum, sumsq, N);

  const int rowtiles = (N + 15) / 16;
  const int qkv_waves = rowtiles * 5;
  ASVT_qkv_kernel<<<(qkv_waves + 7) / 8, 256, 0, stream>>>(feat, Wq, Wk, Wv, q, k, v, N);

  ASVT_attn_kernel<<<NBATCH * TILESMAX, 256, 0, stream>>>(q, k, v, starts, r, N);

  const int tr_waves = rowtiles * 4;
  ASVT_trans_kernel<<<(tr_waves + 7) / 8, 256, 0, stream>>>(r, Wt, t, sum, sumsq, N);

  ASVT_bn_kernel<<<4096, 256, 0, stream>>>(feat, t, sum, sumsq, gamma, beta, out, N);
}